// FasterRCNN_86517821213527
// MI455X (gfx1250) — compile-verified
//
#include <hip/hip_runtime.h>
#include <math.h>

#ifndef __has_builtin
#define __has_builtin(x) 0
#endif

#if __has_builtin(__builtin_amdgcn_global_load_async_to_lds_b128)
#define USE_ASYNC_LDS 1
#else
#define USE_ASYNC_LDS 0
#endif

#define KMAX 64
#define NEG_THRESH 0.3f
#define POS_THRESH 0.7f

#if USE_ASYNC_LDS
typedef int v4i_t __attribute__((ext_vector_type(4)));
typedef __attribute__((address_space(1))) v4i_t gbl_v4i;   // prints as "__device__" in diags
typedef __attribute__((address_space(3))) v4i_t lds_v4i;   // prints as "__shared__"
#endif

// Pure min/max/add/mul/div: no FMA-contraction ambiguity, so pass1 and pass2
// recompute bit-identical IoUs (required for the gt_max equality test).
__device__ __forceinline__ float iou_pair(float ax1, float ay1, float ax2, float ay2, float ba,
                                          float qx1, float qy1, float qx2, float qy2, float qa) {
    float iw = fminf(ax2, qx2) - fmaxf(ax1, qx1) + 1.0f;
    float ih = fminf(ay2, qy2) - fmaxf(ay1, qy1) + 1.0f;
    iw = fmaxf(iw, 0.0f);
    ih = fmaxf(ih, 0.0f);
    float inter = iw * ih;
    return inter / (ba + qa - inter);  // denom >= max(ba,qa) > 0, never NaN
}

__global__ void rpn_init_ws(unsigned int* __restrict__ ws) {
    int t = threadIdx.x;
    if (t < KMAX + 1) ws[t] = 0u;  // [0..63]=gt_max bits (IoU>=0 so +0 bits ok), [64]=count
}

// ---- Pass 1: gt_max_overlaps[k] = max_n IoU(n,k) ------------------------------
__global__ void rpn_pass1_gtmax(const float4* __restrict__ anchors,
                                const float4* __restrict__ gt,
                                unsigned int* __restrict__ gmax,
                                int N, int K) {
    __shared__ float4 s_raw[KMAX];
    __shared__ float s_gx1[KMAX], s_gy1[KMAX], s_gx2[KMAX], s_gy2[KMAX], s_qa[KMAX];
    __shared__ unsigned int s_gmax[KMAX];
    const int t = threadIdx.x;
    const int lane = t & 31;

#if USE_ASYNC_LDS
    if (t < K) {
        __builtin_amdgcn_global_load_async_to_lds_b128(
            (gbl_v4i*)(gt + t), (lds_v4i*)&s_raw[t], 0, 0);
    }
    asm volatile("s_wait_asynccnt 0" ::: "memory");
#else
    if (t < K) s_raw[t] = gt[t];
#endif
    if (t < KMAX) s_gmax[t] = 0u;
    __syncthreads();
    if (t < K) {
        float4 g = s_raw[t];
        s_gx1[t] = g.x; s_gy1[t] = g.y; s_gx2[t] = g.z; s_gy2[t] = g.w;
        s_qa[t] = (g.z - g.x + 1.0f) * (g.w - g.y + 1.0f);
    }
    __syncthreads();

    const int i = blockIdx.x * blockDim.x + t;
    const bool valid = (i < N);
    float4 a = valid ? anchors[i] : make_float4(0.f, 0.f, 0.f, 0.f);
    const float ba = (a.z - a.x + 1.0f) * (a.w - a.y + 1.0f);

    #pragma unroll 4
    for (int k = 0; k < K; ++k) {
        float v = iou_pair(a.x, a.y, a.z, a.w, ba,
                           s_gx1[k], s_gy1[k], s_gx2[k], s_gy2[k], s_qa[k]);
        if (!valid) v = 0.0f;  // IoU >= 0, so padding with 0 never changes the max
        // wave32 max-reduction
        #pragma unroll
        for (int off = 16; off > 0; off >>= 1)
            v = fmaxf(v, __shfl_xor(v, off, 32));
        if (lane == 0) atomicMax(&s_gmax[k], __float_as_uint(v));  // nonneg -> bit order == float order
    }
    __syncthreads();
    if (t < K) atomicMax(&gmax[t], s_gmax[t]);
}

// ---- Pass 2: labels, bbox_targets, inside weights, labeled count --------------
__global__ void rpn_pass2_labels(const float4* __restrict__ anchors,
                                 const float4* __restrict__ gt,
                                 const unsigned int* __restrict__ gmax,
                                 unsigned int* __restrict__ count,
                                 float* __restrict__ out,
                                 int N, int K) {
    __shared__ float4 s_raw[KMAX];
    __shared__ float s_gx1[KMAX], s_gy1[KMAX], s_gx2[KMAX], s_gy2[KMAX], s_qa[KMAX];
    __shared__ unsigned int s_gbits[KMAX];
    const int t = threadIdx.x;
    const int lane = t & 31;

#if USE_ASYNC_LDS
    if (t < K) {
        __builtin_amdgcn_global_load_async_to_lds_b128(
            (gbl_v4i*)(gt + t), (lds_v4i*)&s_raw[t], 0, 0);
    }
    asm volatile("s_wait_asynccnt 0" ::: "memory");
#else
    if (t < K) s_raw[t] = gt[t];
#endif
    if (t < K) s_gbits[t] = gmax[t];
    __syncthreads();
    if (t < K) {
        float4 g = s_raw[t];
        s_gx1[t] = g.x; s_gy1[t] = g.y; s_gx2[t] = g.z; s_gy2[t] = g.w;
        s_qa[t] = (g.z - g.x + 1.0f) * (g.w - g.y + 1.0f);
    }
    __syncthreads();

    const int i = blockIdx.x * blockDim.x + t;
    const bool valid = (i < N);
    float4 a = valid ? anchors[i] : make_float4(0.f, 0.f, 0.f, 0.f);
    const float ba = (a.z - a.x + 1.0f) * (a.w - a.y + 1.0f);

    float best = -1.0f;  // first strictly-greater keeps first occurrence == jnp.argmax
    int arg = 0;
    bool isbest = false;
    #pragma unroll 4
    for (int k = 0; k < K; ++k) {
        float v = iou_pair(a.x, a.y, a.z, a.w, ba,
                           s_gx1[k], s_gy1[k], s_gx2[k], s_gy2[k], s_qa[k]);
        if (v > best) { best = v; arg = k; }
        isbest |= (__float_as_uint(v) == s_gbits[k]);  // exact float == (nonneg, no NaN)
    }

    // label chain: -1 -> 0 (max<0.3) -> 1 (gt best) -> 1 (max>=0.7)
    float lab = -1.0f;
    if (best < NEG_THRESH) lab = 0.0f;
    if (isbest)            lab = 1.0f;
    if (best >= POS_THRESH) lab = 1.0f;

    // one atomic per wave for num_examples (labels >= 0)
    bool pred = valid && (lab >= 0.0f);
    unsigned long long m = __ballot(pred ? 1 : 0);
    if (lane == 0) {
        unsigned int c = (unsigned int)__popcll(m);
        if (c) atomicAdd(count, c);
    }

    if (valid) {
        out[i] = lab;  // labels
        // bbox_transform(anchor, gt[argmax])
        float ew = a.z - a.x + 1.0f, eh = a.w - a.y + 1.0f;
        float ecx = a.x + 0.5f * ew, ecy = a.y + 0.5f * eh;
        float gx1 = s_gx1[arg], gy1 = s_gy1[arg], gx2 = s_gx2[arg], gy2 = s_gy2[arg];
        float gw = gx2 - gx1 + 1.0f, gh = gy2 - gy1 + 1.0f;
        float gcx = gx1 + 0.5f * gw, gcy = gy1 + 0.5f * gh;
        float4 tgt = make_float4((gcx - ecx) / ew, (gcy - ecy) / eh,
                                 logf(gw / ew), logf(gh / eh));
        ((float4*)(out + (size_t)N))[i] = tgt;                       // bbox_targets
        float inw = (lab == 1.0f) ? 1.0f : 0.0f;                     // INSIDE_W = 1,1,1,1
        ((float4*)(out + 5ull * (size_t)N))[i] = make_float4(inw, inw, inw, inw);
    }
}

// ---- Pass 3: outside weights = (label>=0) ? 1/num_examples : 0 ----------------
__global__ void rpn_pass3_outside(const float* __restrict__ labels,
                                  const unsigned int* __restrict__ count,
                                  float* __restrict__ outw,  // points at out + 9N
                                  int N) {
    int i = blockIdx.x * blockDim.x + threadIdx.x;
    if (i >= N) return;
    float w = 1.0f / (float)(*count);  // L2-cached scalar; matches jnp 1/num (inf if 0)
    float lab = labels[i];
    float v = (lab >= 0.0f) ? w : 0.0f;
    ((float4*)outw)[i] = make_float4(v, v, v, v);
}

extern "C" void kernel_launch(void* const* d_in, const int* in_sizes, int n_in,
                              void* d_out, int out_size, void* d_ws, size_t ws_size,
                              hipStream_t stream) {
    const float4* anchors = (const float4*)d_in[0];
    const float4* gt      = (const float4*)d_in[1];
    const int N = in_sizes[0] / 4;   // 200000
    const int K = in_sizes[1] / 4;   // 64 (<= KMAX)

    float* out = (float*)d_out;                 // [labels N | targets 4N | inside 4N | outside 4N]
    unsigned int* ws    = (unsigned int*)d_ws;
    unsigned int* gmax  = ws;                   // 64 slots
    unsigned int* count = ws + KMAX;            // 1 slot

    const int threads = 256;                    // 8 wave32 waves
    const int blocks = (N + threads - 1) / threads;

    rpn_init_ws<<<1, 128, 0, stream>>>(ws);
    rpn_pass1_gtmax<<<blocks, threads, 0, stream>>>(anchors, gt, gmax, N, K);
    rpn_pass2_labels<<<blocks, threads, 0, stream>>>(anchors, gt, gmax, count, out, N, K);
    rpn_pass3_outside<<<blocks, threads, 0, stream>>>(out, count, out + 9ull * (size_t)N, N);
}